// AlignShouldersToXAxis_38525856645650
// MI455X (gfx1250) — compile-verified
//
#include <hip/hip_runtime.h>
#include <math.h>

// ---------------------------------------------------------------------------
// AlignShouldersToXAxis on MI455X (gfx1250)
//
// Bandwidth-bound: 315 MB total traffic, ~0.2 GFLOP -> floor ~13.5 us @23.3TB/s.
// Phase 1: per-batch Rodrigues matrix (128 threads total).
// Phase 2: streaming rotate. Per wave: 4x V_WMMA_F32_16X16X4_F32
//          (A = 16-point tile 16x4, B = R^T padded 4x16, D = rotated points),
//          LDS transpose, then block-coalesced 128-bit non-temporal stores.
// ---------------------------------------------------------------------------

typedef __attribute__((ext_vector_type(2))) float v2f;
typedef __attribute__((ext_vector_type(4))) float f4;
typedef __attribute__((ext_vector_type(8))) float v8f;

#define JOINTS 25
#define FRAMES 4096
#define PTS_PER_BATCH (FRAMES * JOINTS)        // 102400
#define FLOATS_PER_BATCH (PTS_PER_BATCH * 3)   // 307200
#define PTS_PER_BLOCK 512
#define LEFT_SH 5
#define RIGHT_SH 6

// ---------------- Phase 1: one rotation matrix per batch --------------------
__global__ void compute_rotations(const float* __restrict__ in,
                                  float* __restrict__ Rws, int nbatch) {
    int b = blockIdx.x * blockDim.x + threadIdx.x;
    if (b >= nbatch) return;
    const float* s = in + (size_t)b * FLOATS_PER_BATCH;

    // frame 0, joints 5 and 6
    float plx = s[LEFT_SH * 3 + 0],  ply = s[LEFT_SH * 3 + 1];
    float prx = s[RIGHT_SH * 3 + 0], pry = s[RIGHT_SH * 3 + 1];

    float svx = prx - plx, svy = pry - ply;        // z projected to 0
    float sv_norm = sqrtf(svx * svx + svy * svy);
    float inv = 1.0f / fmaxf(sv_norm, 1e-12f);
    float nx = svx * inv, ny = svy * inv, nz = 0.0f;

    // axis = cross(svn, [1,0,0]) = (0, nz, -ny)
    float ax = 0.0f, ay = nz, az = -ny;
    float cosang = fminf(fmaxf(nx, -1.0f), 1.0f);  // dot(svn, target)
    float angle = acosf(cosang);
    float axis_norm = sqrtf(ax * ax + ay * ay + az * az);
    float ainv = 1.0f / fmaxf(axis_norm, 1e-12f);
    float ux = ax * ainv, uy = ay * ainv, uz = az * ainv;

    float sn = sinf(angle), cs = cosf(angle);
    float oc = 1.0f - cs;

    // R = I + sin*K + (1-cos)*K^2  (unit axis closed form)
    float R[9];
    R[0] = cs + oc * ux * ux;      R[1] = oc * ux * uy - sn * uz; R[2] = oc * ux * uz + sn * uy;
    R[3] = oc * uy * ux + sn * uz; R[4] = cs + oc * uy * uy;      R[5] = oc * uy * uz - sn * ux;
    R[6] = oc * uz * ux - sn * uy; R[7] = oc * uz * uy + sn * ux; R[8] = cs + oc * uz * uz;

    bool valid = (sv_norm >= 1e-6f) && (axis_norm >= 1e-6f) && !(angle != angle);
    if (!valid) {
        R[0] = 1.0f; R[1] = 0.0f; R[2] = 0.0f;
        R[3] = 0.0f; R[4] = 1.0f; R[5] = 0.0f;
        R[6] = 0.0f; R[7] = 0.0f; R[8] = 1.0f;
    }
    float* dst = Rws + b * 12;   // stride 12 floats (48B aligned)
    #pragma unroll
    for (int i = 0; i < 9; ++i) dst[i] = R[i];
}

// ---------------- Phase 2: WMMA streaming rotate ----------------------------
// D = A x B:  A[i][k] = point_i component k (k<3, col 3 = 0)  -> 16x4
//             B[k][j] = R[j][k] (R^T, zero padded)            -> 4x16
//             D[i][j] = (R * p_i)[j]                          -> 16x16, cols 0..2 used
__global__ __launch_bounds__(256)
void rotate_points_wmma(const float* __restrict__ in,
                        const float* __restrict__ Rws,
                        float* __restrict__ out) {
    __shared__ __align__(16) float lds[PTS_PER_BLOCK * 3];   // 6 KB

    const int b = blockIdx.y;
    const size_t base = (size_t)b * FLOATS_PER_BATCH + (size_t)blockIdx.x * (PTS_PER_BLOCK * 3);
    const float* __restrict__ inB = in + base;
    const float* __restrict__ Rb = Rws + b * 12;

    const int tid  = threadIdx.x;
    const int wave = tid >> 5;
    const int lane = tid & 31;
    const int l    = lane & 15;
    const bool low = (lane < 16);

    // ---- Build B operand: 4x16 f32 in 2 VGPRs.
    // Layout (mirrors documented 16x4 A layout): VGPR0 lanes0-15 = row K=0,
    // lanes16-31 = row K=2; VGPR1 lanes0-15 = K=1, lanes16-31 = K=3.
    // B[k][col] = R[col][k] for k<3 && col<3, else 0.
    const int col  = l;
    const int row0 = low ? 0 : 2;   // row for .x ; row for .y is row0+1
    v2f bmat;
    bmat.x = (col < 3) ? Rb[col * 3 + row0] : 0.0f;                  // row0 in {0,2} < 3
    bmat.y = (col < 3 && (row0 + 1) < 3) ? Rb[col * 3 + row0 + 1] : 0.0f;

    const int wavePt = wave * 64;   // 64 points per wave, 4 WMMA tiles of 16

    // Branch-free A-operand addressing (documented 32-bit 16x4 A layout:
    // lanes0-15: VGPR0=K0(x), VGPR1=K1(y); lanes16-31: VGPR0=K2(z), VGPR1=K3(0)).
    //  .x : per-lane address selects x (low half) or z (high half) -> 1 load
    //  .y : ALL lanes load y, then v_cndmask zeroes the high half   -> 1 load + select
    const float* __restrict__ pX = inB + (wavePt + l) * 3 + (low ? 0 : 2);
    const float* __restrict__ pY = inB + (wavePt + l) * 3 + 1;
    // LDS scatter base (used only by lanes with l < 3)
    float* __restrict__ ldsBase = lds + (wavePt + (low ? 0 : 8)) * 3 + l;

    #pragma unroll
    for (int t = 0; t < 4; ++t) {
        const int fo = t * 48;      // 16 points * 3 floats per tile

        v2f amat;
        amat.x = __builtin_nontemporal_load(pX + fo);
        float ycomp = __builtin_nontemporal_load(pY + fo);
        amat.y = low ? ycomp : 0.0f;

        v8f acc = {};
        acc = __builtin_amdgcn_wmma_f32_16x16x4_f32(
            /*neg_a=*/false, amat, /*neg_b=*/false, bmat,
            /*c_mod=*/(short)0, acc, /*reuse_a=*/false, /*reuse_b=*/false);

        // ---- Scatter D cols 0..2 to LDS (documented C/D layout:
        // VGPR v: lanes0-15 -> row M=v, lanes16-31 -> row M=v+8; col = lane%16)
        if (l < 3) {
            float* __restrict__ d = ldsBase + fo;
            d[0]  = acc[0];
            d[3]  = acc[1];
            d[6]  = acc[2];
            d[9]  = acc[3];
            d[12] = acc[4];
            d[15] = acc[5];
            d[18] = acc[6];
            d[21] = acc[7];
        }
    }

    __syncthreads();

    // ---- Block-coalesced writeback: 512 pts * 3 = 1536 floats = 384 float4.
    // Stream is written once and never re-read -> non-temporal 128-bit stores.
    f4* __restrict__ outV = (f4*)(out + base);
    const f4* __restrict__ ldsV = (const f4*)lds;
    __builtin_nontemporal_store(ldsV[tid], &outV[tid]);
    if (tid < 128)
        __builtin_nontemporal_store(ldsV[256 + tid], &outV[256 + tid]);
}

// ---------------------------------------------------------------------------
extern "C" void kernel_launch(void* const* d_in, const int* in_sizes, int n_in,
                              void* d_out, int out_size, void* d_ws, size_t ws_size,
                              hipStream_t stream) {
    const float* seq = (const float*)d_in[0];
    float* out = (float*)d_out;
    float* Rws = (float*)d_ws;                       // nbatch * 12 floats

    const int nbatch = in_sizes[0] / FLOATS_PER_BATCH;   // 128

    compute_rotations<<<(nbatch + 127) / 128, 128, 0, stream>>>(seq, Rws, nbatch);

    dim3 grid(PTS_PER_BATCH / PTS_PER_BLOCK, nbatch);    // (200, 128)
    rotate_points_wmma<<<grid, 256, 0, stream>>>(seq, Rws, out);
}